// AttOutput_17695265260104
// MI455X (gfx1250) — compile-verified
//
#include <hip/hip_runtime.h>
#include <cstddef>

typedef __attribute__((ext_vector_type(2))) float v2f;
typedef __attribute__((ext_vector_type(8))) float v8f;

#define B_N 2048
#define T_N 200
#define D_N 256
#define C_N 50
#define C_PAD 64
#define EPSF 1e-7f

// ---------------------------------------------------------------------------
// Kernel 0: pad + transpose W2 [256,50] -> W2pT [64,256] (cols 50..63 = 0).
// Makes the WMMA B-fragment loads unconditional and K-contiguous (b64).
// Runs once over 64 KB -- negligible next to the 420 MB x stream.
// ---------------------------------------------------------------------------
__global__ __launch_bounds__(256) void pad_w2_kernel(
    const float* __restrict__ W2, float* __restrict__ W2pT) {
  const int i = blockIdx.x * 256 + threadIdx.x;   // i < 64*256
  const int n = i >> 8;        // padded class index 0..63
  const int k = i & 255;       // feature index 0..255
  W2pT[n * D_N + k] = (n < C_N) ? W2[k * C_N + n] : 0.f;
}

// ---------------------------------------------------------------------------
// Kernel 1: fused additive attention + pooling. One workgroup per batch row.
//   e_t = tanh(x[b,t,:]·W1 + b[t]);  a = exp(e)/(sum+eps);  c[b,:] = a·x[b]
// Memory-bound on x (420 MB); x[b] is 200 KB so the pooling-pass re-read
// hits the 192 MB L2.
// ---------------------------------------------------------------------------
__global__ __launch_bounds__(256) void attn_pool_kernel(
    const float* __restrict__ x, const float* __restrict__ W1,
    const float* __restrict__ bvec, float* __restrict__ c) {
  __shared__ float s_a[T_N];
  __shared__ float s_red[8];
  __shared__ float s_total;

  const int b = blockIdx.x;
  const int tid = threadIdx.x;
  const int lane = tid & 31;
  const int wave = tid >> 5;

  // Each lane owns 8 contiguous channels of W1 (two b128 loads).
  const float4* w1p = reinterpret_cast<const float4*>(W1 + lane * 8);
  const float4 w1a = w1p[0];
  const float4 w1b = w1p[1];

  const float* xb = x + (size_t)b * T_N * D_N;

  // Phase A: energies. 8 waves x 25 timesteps. Per t: lane-parallel dot over
  // D=256 (two b128 loads/lane) + 5-step wave32 xor reduction.
  for (int i = 0; i < 25; ++i) {
    const int t = wave * 25 + i;
    const float4* px = reinterpret_cast<const float4*>(xb + t * D_N + lane * 8);
    const float4 xa = px[0];
    const float4 xc = px[1];
    float p = xa.x * w1a.x + xa.y * w1a.y + xa.z * w1a.z + xa.w * w1a.w +
              xc.x * w1b.x + xc.y * w1b.y + xc.z * w1b.z + xc.w * w1b.w;
    for (int off = 16; off > 0; off >>= 1) p += __shfl_xor(p, off, 32);
    if (lane == 0) s_a[t] = tanhf(p + bvec[t]);
  }
  __syncthreads();

  // Phase B: exp + block-wide sum + normalize in LDS.
  float part = 0.f;
  for (int i = tid; i < T_N; i += 256) {
    const float ex = __expf(s_a[i]);
    s_a[i] = ex;
    part += ex;
  }
  for (int off = 16; off > 0; off >>= 1) part += __shfl_xor(part, off, 32);
  if (lane == 0) s_red[wave] = part;
  __syncthreads();
  if (tid == 0) {
    float tot = 0.f;
    for (int w = 0; w < 8; ++w) tot += s_red[w];
    s_total = 1.0f / (tot + EPSF);
  }
  __syncthreads();
  const float inv = s_total;
  for (int i = tid; i < T_N; i += 256) s_a[i] *= inv;
  __syncthreads();

  // Phase C: c[d] = sum_t a_t * x[b,t,d]. tid == d; fully coalesced rows.
  float acc = 0.f;
  for (int t = 0; t < T_N; ++t) acc += s_a[t] * xb[t * D_N + tid];
  c[(size_t)b * D_N + tid] = acc;
}

// ---------------------------------------------------------------------------
// Kernel 2: o = rownorm(exp(c @ W2)) via V_WMMA_F32_16X16X4_F32.
// M=2048 (16 rows/WG), K=256 (64 wmma issues), N padded 50->64 (one 16-col
// tile per wave, 4 waves). Inner loop is branch-free: ds_load_b64 (A) +
// global b64 (B from padded W2^T) + v_wmma.
// ---------------------------------------------------------------------------
#define SA_STRIDE 260   // 16x256 A tile, padded: stride 256 floats = exact
                        // multiple of 64 banks -> 16-way conflict without pad
#define SE_STRIDE 68

__global__ __launch_bounds__(128) void classify_wmma_kernel(
    const float* __restrict__ c, const float* __restrict__ W2pT,
    float* __restrict__ out) {
  __shared__ float sA[16 * SA_STRIDE];
  __shared__ float sE[16 * SE_STRIDE];
  __shared__ float sInv[16];

  const int tid = threadIdx.x;
  const int lane = tid & 31;
  const int wave = tid >> 5;
  const int m0 = blockIdx.x * 16;

  // Stage the 16x256 slice of c into LDS (coalesced, shared by all 4 waves).
  for (int i = tid; i < 16 * D_N; i += 128) {
    const int r = i >> 8;
    const int col = i & 255;
    sA[r * SA_STRIDE + col] = c[(size_t)(m0 + r) * D_N + col];
  }
  __syncthreads();

  const int n0 = wave * 16;          // this wave's 16-column N tile
  const int half = lane >> 4;        // lane half selects K pair
  const int arow = lane & 15;        // A: matrix row; B: matrix column
  const int nIdx = n0 + arow;        // always valid in padded W2^T

  const float* aRow = &sA[arow * SA_STRIDE];
  const float* bCol = &W2pT[(size_t)nIdx * D_N];

  // ISA 32-bit 16x4 fragment layout: VGPR v, lane-half h -> K = 2h + v.
  v8f acc = {};
  #pragma unroll 4
  for (int k = 0; k < D_N; k += 4) {
    const int kk = k + 2 * half;
    v2f a, bm;
    a[0] = aRow[kk];
    a[1] = aRow[kk + 1];
    bm[0] = bCol[kk];
    bm[1] = bCol[kk + 1];
    acc = __builtin_amdgcn_wmma_f32_16x16x4_f32(
        /*neg_a=*/false, a, /*neg_b=*/false, bm,
        /*c_mod=*/(short)0, acc, /*reuse_a=*/false, /*reuse_b=*/false);
  }

  // C/D layout: VGPR r, lane-half h -> row = r + 8h, col = lane&15.
  #pragma unroll
  for (int r = 0; r < 8; ++r) {
    const int row = r + 8 * half;
    sE[row * SE_STRIDE + n0 + arow] = __expf(acc[r]);
  }
  __syncthreads();

  if (tid < 16) {
    float s = 0.f;
    for (int j = 0; j < C_N; ++j) s += sE[tid * SE_STRIDE + j];
    sInv[tid] = 1.0f / (s + EPSF);
  }
  __syncthreads();

  for (int i = tid; i < 16 * C_N; i += 128) {
    const int row = i / C_N;
    const int col = i - row * C_N;
    out[(size_t)(m0 + row) * C_N + col] = sE[row * SE_STRIDE + col] * sInv[row];
  }
}

// ---------------------------------------------------------------------------
extern "C" void kernel_launch(void* const* d_in, const int* in_sizes, int n_in,
                              void* d_out, int out_size, void* d_ws, size_t ws_size,
                              hipStream_t stream) {
  (void)in_sizes; (void)n_in; (void)out_size; (void)ws_size;
  const float* x    = (const float*)d_in[0];  // [2048, 200, 256]
  const float* W1   = (const float*)d_in[1];  // [256]
  const float* bvec = (const float*)d_in[2];  // [200]
  const float* W2   = (const float*)d_in[3];  // [256, 50]
  float* out = (float*)d_out;                 // [2048, 50]

  float* c    = (float*)d_ws;                           // 2048*256*4 = 2 MB
  float* W2pT = (float*)((char*)d_ws + (size_t)B_N * D_N * sizeof(float)); // 64 KB

  pad_w2_kernel<<<(C_PAD * D_N) / 256, 256, 0, stream>>>(W2, W2pT);
  attn_pool_kernel<<<B_N, 256, 0, stream>>>(x, W1, bvec, c);
  classify_wmma_kernel<<<B_N / 16, 128, 0, stream>>>(c, W2pT, out);
}